// LlamaDecoderLayer_63376537419811
// MI455X (gfx1250) — compile-verified
//
#include <hip/hip_runtime.h>

typedef float v2f __attribute__((ext_vector_type(2)));
typedef float v8f __attribute__((ext_vector_type(8)));

#define WMMA_F32(a, b, c) \
  __builtin_amdgcn_wmma_f32_16x16x4_f32(false, (a), false, (b), (short)0, (c), false, false)

constexpr int S_   = 2048;
constexpr int HID  = 4096;
constexpr int NH   = 32;
constexpr int NKV  = 8;
constexpr int HD   = 128;      // HID / NH
constexpr int G    = NH / NKV; // 4
constexpr int INTER = 14336;
constexpr float EPS = 1e-5f;
constexpr float THETA = 10000.0f;

// ---------------------------------------------------------------- RMSNorm ---
__global__ __launch_bounds__(256) void rmsnorm_kernel(const float* __restrict__ x,
                                                      const float* __restrict__ w,
                                                      float* __restrict__ out) {
  const int row = blockIdx.x;
  const float* xr = x + (size_t)row * HID;
  float ss = 0.f;
  for (int i = threadIdx.x; i < HID; i += 256) { float v = xr[i]; ss += v * v; }
#pragma unroll
  for (int off = 16; off >= 1; off >>= 1) ss += __shfl_xor(ss, off, 32);
  __shared__ float red[8];
  __shared__ float tot;
  if ((threadIdx.x & 31) == 0) red[threadIdx.x >> 5] = ss;
  __syncthreads();
  if (threadIdx.x == 0) {
    float t = 0.f;
#pragma unroll
    for (int i = 0; i < 8; ++i) t += red[i];
    tot = rsqrtf(t / (float)HID + EPS);
  }
  __syncthreads();
  const float rr = tot;
  float* orow = out + (size_t)row * HID;
  for (int i = threadIdx.x; i < HID; i += 256) orow[i] = xr[i] * rr * w[i];
}

// ------------------------------------------------------------- NeoX RoPE ---
__global__ void rope_kernel(float* __restrict__ q, float* __restrict__ k,
                            const int* __restrict__ pos) {
  const int s = blockIdx.x;
  const int h = blockIdx.y;  // 0..NH+NKV-1
  const int d = threadIdx.x; // 0..63 (half = HD/2)
  float* p;
  if (h < NH) p = q + (size_t)s * (NH * HD) + (size_t)h * HD;
  else        p = k + (size_t)s * (NKV * HD) + (size_t)(h - NH) * HD;
  const float fp  = (float)pos[s];
  const float inv = __powf(THETA, -2.0f * (float)d / (float)HD);
  const float ang = fp * inv;
  const float c = __cosf(ang), sn = __sinf(ang);
  const float x1 = p[d], x2 = p[d + 64];
  p[d]      = x1 * c - x2 * sn;
  p[d + 64] = x2 * c + x1 * sn;
}

// ------------------------------------------------ Tiled WMMA fp32 GEMM -----
// C(M,N) = A(M,K) @ B(K,N)   row-major everywhere.
// MODE 0: plain store.  MODE 1: += res.  MODE 2: silu(C_B) * C_B2 (gate/up).
// Grid: x = M-tiles (fast axis) so consecutive blocks share one B stripe
// (keeps the >L2-sized weight matrices streamed from HBM exactly once while
// the L2-resident activations are re-read from cache).
template <int MODE>
__global__ __launch_bounds__(256) void gemm_wmma_kernel(
    const float* __restrict__ A, const float* __restrict__ B,
    const float* __restrict__ B2, const float* __restrict__ res,
    float* __restrict__ C, int M, int N, int K) {
  constexpr int BM = 64, BN = 64, BK = 16;
  __shared__ __align__(16) float As[BM][20];   // pitch 80B: 16B mult, conflict-free frag reads
  __shared__ __align__(16) float Bs[BK][68];   // pitch 272B
  __shared__ __align__(16) float Bs2[MODE == 2 ? BK : 1][MODE == 2 ? 68 : 1];

  const int tid  = threadIdx.x;
  const int lane = tid & 31;
  const int wave = tid >> 5;
  const int wm = wave & 3;        // 4 row-tiles of 16
  const int wn = wave >> 2;       // 2 col-tiles of 32
  const int lm = lane & 15;
  const int lk = (lane >> 4) * 2; // K half-split per A/B fragment layout

  const int mBlk = blockIdx.x * BM;   // M is the fast grid axis
  const int nBlk = blockIdx.y * BN;

  v8f c0 = {}, c1 = {}, u0 = {}, u1 = {};

  const int ar = tid >> 2, ac = (tid & 3) * 4;   // A coop load: 64 rows x 16 cols
  const int br = tid >> 4, bc = (tid & 15) * 4;  // B coop load: 16 rows x 64 cols

  const float* Arow = A + (size_t)(mBlk + ar) * K + ac;
  // software pipeline: preload first K-tile into registers
  float4 av = *(const float4*)(Arow);
  float4 bv = *(const float4*)(B + (size_t)br * N + nBlk + bc);
  float4 bv2 = {};
  if (MODE == 2) bv2 = *(const float4*)(B2 + (size_t)br * N + nBlk + bc);

  for (int kb = 0; kb < K; kb += BK) {
    *(float4*)(&As[ar][ac]) = av;
    *(float4*)(&Bs[br][bc]) = bv;
    if (MODE == 2) *(float4*)(&Bs2[br][bc]) = bv2;
    __syncthreads();
    if (kb + BK < K) { // issue next tile's loads; they overlap the WMMAs below
      av = *(const float4*)(Arow + kb + BK);
      bv = *(const float4*)(B + (size_t)(kb + BK + br) * N + nBlk + bc);
      if (MODE == 2) bv2 = *(const float4*)(B2 + (size_t)(kb + BK + br) * N + nBlk + bc);
    }
#pragma unroll
    for (int kk = 0; kk < 4; ++kk) {
      const int k0 = kk * 4 + lk;
      v2f a = *(const v2f*)(&As[wm * 16 + lm][k0]);
      v2f b0, b1;
      b0.x = Bs[k0][wn * 32 + lm];        b0.y = Bs[k0 + 1][wn * 32 + lm];
      b1.x = Bs[k0][wn * 32 + 16 + lm];   b1.y = Bs[k0 + 1][wn * 32 + 16 + lm];
      c0 = WMMA_F32(a, b0, c0);
      c1 = WMMA_F32(a, b1, c1);
      if (MODE == 2) {
        v2f d0, d1;
        d0.x = Bs2[k0][wn * 32 + lm];      d0.y = Bs2[k0 + 1][wn * 32 + lm];
        d1.x = Bs2[k0][wn * 32 + 16 + lm]; d1.y = Bs2[k0 + 1][wn * 32 + 16 + lm];
        u0 = WMMA_F32(a, d0, u0);
        u1 = WMMA_F32(a, d1, u1);
      }
    }
    __syncthreads();
  }

  // C layout: row = r + 8*(lane>=16), col = lane%16
  const int rowBase = mBlk + wm * 16 + (lane >> 4) * 8;
  const int colBase = nBlk + wn * 32 + lm;
#pragma unroll
  for (int r = 0; r < 8; ++r) {
    const size_t i0 = (size_t)(rowBase + r) * N + colBase;
    const size_t i1 = i0 + 16;
    if (MODE == 0) {
      C[i0] = c0[r];
      C[i1] = c1[r];
    } else if (MODE == 1) {
      C[i0] = c0[r] + res[i0];
      C[i1] = c1[r] + res[i1];
    } else {
      const float g0 = c0[r], g1 = c1[r];
      C[i0] = (g0 / (1.f + __expf(-g0))) * u0[r];
      C[i1] = (g1 / (1.f + __expf(-g1))) * u1[r];
    }
  }
}

// -------------------------------------- Flash causal GQA attention (fp32) --
// grid = (NH, S/128), 256 thr = 8 waves; each wave owns a 16-row q tile.
__global__ __launch_bounds__(256) void attn_kernel(
    const float* __restrict__ q, const float* __restrict__ k,
    const float* __restrict__ v, float* __restrict__ attn) {
  __shared__ float P[8][16][17]; // per-wave C-layout -> A-fragment staging
  const int lane  = threadIdx.x & 31;
  const int wave  = threadIdx.x >> 5;
  const int lm    = lane & 15;
  const int lhalf = lane >> 4;
  const int lk    = lhalf * 2;

  const int h    = blockIdx.x;
  const int kvh  = h / G;
  const int qbase = blockIdx.y * 128 + wave * 16;

  const float* Qh = q + (size_t)h * HD;
  const float* Kh = k + (size_t)kvh * HD;
  const float* Vh = v + (size_t)kvh * HD;
  constexpr int QS = NH * HD;   // 4096
  constexpr int KS = NKV * HD;  // 1024
  const float scale = 0.088388347648318447f; // 1/sqrt(HD)

  // Q fragments are invariant across kv tiles: hoist into registers (64 VGPRs)
  v2f qa[HD / 4];
#pragma unroll
  for (int kk = 0; kk < HD / 4; ++kk)
    qa[kk] = *(const v2f*)(Qh + (size_t)(qbase + lm) * QS + kk * 4 + lk);

  v8f o[8];
#pragma unroll
  for (int t = 0; t < 8; ++t) o[t] = {};
  float mrow[8], lrow[8];
#pragma unroll
  for (int r = 0; r < 8; ++r) { mrow[r] = -3.0e38f; lrow[r] = 0.f; }

  const int rowQ   = qbase + lhalf * 8; // + r gives this lane's C-layout row
  const int ntiles = qbase / 16 + 1;    // causal: kv tiles with kbase <= qbase

  for (int kt = 0; kt < ntiles; ++kt) {
    const int kbase = kt * 16;
    // scores tile = Q(16xHD) . K^T(HDx16), K-dim stepped by 4
    v8f sc = {};
#pragma unroll
    for (int kk = 0; kk < HD / 4; ++kk) {
      v2f b = *(const v2f*)(Kh + (size_t)(kbase + lm) * KS + kk * 4 + lk);
      sc = WMMA_F32(qa[kk], b, sc);
    }
    const int colK = kbase + lm;
    float p[8], tmax[8];
#pragma unroll
    for (int r = 0; r < 8; ++r) {
      float x = sc[r] * scale;
      if (colK > rowQ + r) x = -3.0e38f; // causal mask
      p[r] = x;
      float mx = x;
#pragma unroll
      for (int off = 1; off < 16; off <<= 1) mx = fmaxf(mx, __shfl_xor(mx, off, 16));
      tmax[r] = mx;
    }
    float alpha[8];
#pragma unroll
    for (int r = 0; r < 8; ++r) {
      const float mn = fmaxf(mrow[r], tmax[r]);
      alpha[r] = __expf(mrow[r] - mn);
      const float e = __expf(p[r] - mn);
      p[r] = e;
      float s = e;
#pragma unroll
      for (int off = 1; off < 16; off <<= 1) s += __shfl_xor(s, off, 16);
      lrow[r] = lrow[r] * alpha[r] + s;
      mrow[r] = mn;
    }
#pragma unroll
    for (int t = 0; t < 8; ++t)
#pragma unroll
      for (int r = 0; r < 8; ++r) o[t][r] *= alpha[r];

    // C-layout -> LDS (row-major P), then reload as A fragments
#pragma unroll
    for (int r = 0; r < 8; ++r) P[wave][lhalf * 8 + r][lm] = p[r];
    asm volatile("s_wait_dscnt 0x0" ::: "memory"); // per-wave DS in-order: stores done
#pragma unroll
    for (int kk = 0; kk < 4; ++kk) {
      const int k0 = kk * 4 + lk;
      v2f pa;
      pa.x = P[wave][lm][k0];
      pa.y = P[wave][lm][k0 + 1];
#pragma unroll
      for (int t = 0; t < 8; ++t) {
        v2f vb;
        vb.x = Vh[(size_t)(kbase + k0) * KS + t * 16 + lm];
        vb.y = Vh[(size_t)(kbase + k0 + 1) * KS + t * 16 + lm];
        o[t] = WMMA_F32(pa, vb, o[t]);
      }
    }
    asm volatile("" ::: "memory");
  }

#pragma unroll
  for (int r = 0; r < 8; ++r) {
    const float inv = 1.0f / lrow[r];
    float* orow = attn + (size_t)(rowQ + r) * (NH * HD) + (size_t)h * HD;
#pragma unroll
    for (int t = 0; t < 8; ++t) orow[t * 16 + lm] = o[t][r] * inv;
  }
}

// ---------------------------------------------------------------- driver ---
extern "C" void kernel_launch(void* const* d_in, const int* in_sizes, int n_in,
                              void* d_out, int out_size, void* d_ws, size_t ws_size,
                              hipStream_t stream) {
  const int*   pos = (const int*)d_in[0];
  const float* hid = (const float*)d_in[1];
  const float* wq  = (const float*)d_in[2];
  const float* wk  = (const float*)d_in[3];
  const float* wv  = (const float*)d_in[4];
  const float* wo  = (const float*)d_in[5];
  const float* wg  = (const float*)d_in[6];
  const float* wu  = (const float*)d_in[7];
  const float* wd  = (const float*)d_in[8];
  const float* ln1 = (const float*)d_in[9];
  const float* ln2 = (const float*)d_in[10];

  float* out    = (float*)d_out;
  float* outH   = out;                        // (S, HID) MLP output
  float* outRes = out + (size_t)S_ * HID;     // (S, HID) residual after attn

  float* ws = (float*)d_ws;
  size_t off = 0;
  float* hnorm = ws + off; off += (size_t)S_ * HID;       // rmsnorm1 out
  float* qb    = ws + off; off += (size_t)S_ * NH * HD;   // Q
  float* kb    = ws + off; off += (size_t)S_ * NKV * HD;  // K
  float* vb    = ws + off; off += (size_t)S_ * NKV * HD;  // V
  float* attn  = ws + off; off += (size_t)S_ * NH * HD;   // attn out
  float* h2    = ws + off; off += (size_t)S_ * HID;       // rmsnorm2 out
  float* act   = ws + off;                                // silu(gate)*up

  const dim3 blk(256);

  // 1) h = rmsnorm(x) * ln1
  rmsnorm_kernel<<<S_, blk, 0, stream>>>(hid, ln1, hnorm);
  // 2) Q/K/V projections   (grid.x = M-tiles)
  gemm_wmma_kernel<0><<<dim3(S_ / 64, (NH * HD) / 64), blk, 0, stream>>>(
      hnorm, wq, nullptr, nullptr, qb, S_, NH * HD, HID);
  gemm_wmma_kernel<0><<<dim3(S_ / 64, (NKV * HD) / 64), blk, 0, stream>>>(
      hnorm, wk, nullptr, nullptr, kb, S_, NKV * HD, HID);
  gemm_wmma_kernel<0><<<dim3(S_ / 64, (NKV * HD) / 64), blk, 0, stream>>>(
      hnorm, wv, nullptr, nullptr, vb, S_, NKV * HD, HID);
  // 3) RoPE (NeoX) on q and k
  rope_kernel<<<dim3(S_, NH + NKV), dim3(64), 0, stream>>>(qb, kb, pos);
  // 4) causal GQA attention
  attn_kernel<<<dim3(NH, S_ / 128), blk, 0, stream>>>(qb, kb, vb, attn);
  // 5) residual = attn @ wo + x   (second output)
  gemm_wmma_kernel<1><<<dim3(S_ / 64, HID / 64), blk, 0, stream>>>(
      attn, wo, nullptr, hid, outRes, S_, HID, NH * HD);
  // 6) h2 = rmsnorm(residual) * ln2
  rmsnorm_kernel<<<S_, blk, 0, stream>>>(outRes, ln2, h2);
  // 7) act = silu(h2 @ wg) * (h2 @ wu)   (fused, shared A fragments)
  gemm_wmma_kernel<2><<<dim3(S_ / 64, INTER / 64), blk, 0, stream>>>(
      h2, wg, wu, nullptr, act, S_, INTER, HID);
  // 8) h = act @ wd   (first output)
  gemm_wmma_kernel<0><<<dim3(S_ / 64, HID / 64), blk, 0, stream>>>(
      act, wd, nullptr, nullptr, outH, S_, HID, INTER);
}